// RESCAL_11304353923483
// MI455X (gfx1250) — compile-verified
//
#include <hip/hip_runtime.h>
#include <hip/hip_bf16.h>

typedef __attribute__((ext_vector_type(2))) float v2f;
typedef __attribute__((ext_vector_type(8))) float v8f;

#define BATCH    16384
#define NREL     200
#define DIM      128
#define TILE     16
#define MAXTILES (BATCH / TILE + NREL)   // 1224 worst case (each group adds <=1 partial tile)
#define LSTRIDE  17                      // padded LDS row stride: kills 16-way bank conflicts

// ---------------- setup kernels ----------------

__global__ void k_zero(int* counts, int* ntiles) {
    int i = blockIdx.x * blockDim.x + threadIdx.x;
    if (i < NREL) counts[i] = 0;
    if (i == 0) *ntiles = 0;
}

// ||R_g||^2 for each relation row (16384 fp32 elements each)
__global__ void k_relnorm(const float* __restrict__ rel, float* __restrict__ relnorm) {
    __shared__ float red[256];
    int g = blockIdx.x;
    const float* row = rel + (size_t)g * (DIM * DIM);
    float s = 0.f;
    for (int i = threadIdx.x; i < DIM * DIM; i += 256) {
        float v = row[i];
        s += v * v;
    }
    red[threadIdx.x] = s;
    __syncthreads();
    for (int off = 128; off > 0; off >>= 1) {
        if (threadIdx.x < off) red[threadIdx.x] += red[threadIdx.x + off];
        __syncthreads();
    }
    if (threadIdx.x == 0) relnorm[g] = red[0];
}

__global__ void k_hist(const int* __restrict__ r, int* counts) {
    int i = blockIdx.x * blockDim.x + threadIdx.x;
    if (i < BATCH) atomicAdd(&counts[r[i]], 1);
}

// serial scan over 200 bins + tile-table build (trivial work, one thread)
__global__ void k_scan(const int* __restrict__ counts, int* offsets,
                       int* tg, int* ts, int* tc, int* ntiles) {
    if (threadIdx.x == 0 && blockIdx.x == 0) {
        int pos = 0, t = 0;
        for (int g = 0; g < NREL; ++g) {
            offsets[g] = pos;
            int n = counts[g];
            for (int s = 0; s < n; s += TILE) {
                tg[t] = g;
                ts[t] = pos + s;
                tc[t] = (n - s < TILE) ? (n - s) : TILE;
                ++t;
            }
            pos += n;
        }
        *ntiles = t;
    }
}

// stable (deterministic) counting-sort scatter: rank = #earlier items w/ same relation
__global__ void k_scatter(const int* __restrict__ r, const int* __restrict__ offsets,
                          int* __restrict__ sorted) {
    int i = blockIdx.x * blockDim.x + threadIdx.x;
    if (i >= BATCH) return;
    int g = r[i];
    int rank = 0;
    for (int j = 0; j < i; ++j) rank += (r[j] == g);   // r[] is tiny (64KB), L2-hot
    sorted[offsets[g] + rank] = i;
}

// ---------------- main grouped-GEMM scoring kernel ----------------
// One 4-wave block per 16-item relation tile.
// TR(128x16) = R(128x128) * T(128x16) via v_wmma_f32_16x16x4_f32, for pos & neg tails.
__global__ void __launch_bounds__(128)
k_score(const int* __restrict__ h, const int* __restrict__ pos_t,
        const int* __restrict__ neg_t,
        const float* __restrict__ ent, const float* __restrict__ rel,
        const int* __restrict__ tg, const int* __restrict__ ts, const int* __restrict__ tc,
        const int* __restrict__ ntiles, const int* __restrict__ sorted,
        const float* __restrict__ relnorm, float* __restrict__ loss) {
    __shared__ float sh_h [DIM * LSTRIDE];
    __shared__ float sh_tp[DIM * LSTRIDE];
    __shared__ float sh_tn[DIM * LSTRIDE];
    __shared__ float sh_sp[4][TILE];
    __shared__ float sh_sn[4][TILE];

    int tile = blockIdx.x;
    if (tile >= *ntiles) return;                 // uniform per block
    int g = tg[tile], start = ts[tile], cnt = tc[tile];

    int tid = threadIdx.x;

    // ---- stage 16-item entity tiles into LDS, k-major with padded stride ----
    for (int p = tid; p < TILE * DIM; p += 128) {
        int n = p >> 7;                          // item slot 0..15
        int k = p & (DIM - 1);                   // element 0..127
        int slot = (n < cnt) ? n : 0;            // clamp padded slots (scores unused)
        int item = sorted[start + slot];
        long hb = (long)h[item]     * DIM;
        long pb = (long)pos_t[item] * DIM;
        long nb = (long)neg_t[item] * DIM;
        sh_h [k * LSTRIDE + n] = ent[hb + k];
        sh_tp[k * LSTRIDE + n] = ent[pb + k];
        sh_tn[k * LSTRIDE + n] = ent[nb + k];
    }
    __syncthreads();

    // ---- fp32 WMMA: A = R row-tile (16x4), B = tail tile (4x16) ----
    int wave = tid >> 5;
    int lane = tid & 31;
    int m    = lane & 15;                        // A-row within tile / B,D column (item)
    int hi   = lane >> 4;                        // half-wave: K pairs {0,1} vs {2,3}
    const float* Rg = rel + (size_t)g * (DIM * DIM);
    float sp = 0.f, sn = 0.f;
    for (int rbi = 0; rbi < 2; ++rbi) {
        int rb = wave + 4 * rbi;                 // output-row block 0..7
        const float* Arow = Rg + (size_t)(rb * 16 + m) * DIM;
        v8f accp = {0.f,0.f,0.f,0.f,0.f,0.f,0.f,0.f};
        v8f accn = {0.f,0.f,0.f,0.f,0.f,0.f,0.f,0.f};
        #pragma unroll 4
        for (int kc = 0; kc < DIM; kc += 4) {
            int kk = kc + 2 * hi;
            v2f a = *(const v2f*)(Arow + kk);    // A frag: R[d, kk], R[d, kk+1]
            v2f bp, bn;                          // B frag: T[kk,n], T[kk+1,n]
            bp.x = sh_tp[(kk    ) * LSTRIDE + m];
            bp.y = sh_tp[(kk + 1) * LSTRIDE + m];
            bn.x = sh_tn[(kk    ) * LSTRIDE + m];
            bn.y = sh_tn[(kk + 1) * LSTRIDE + m];
            accp = __builtin_amdgcn_wmma_f32_16x16x4_f32(false, a, false, bp,
                                                         (short)0, accp, false, false);
            accn = __builtin_amdgcn_wmma_f32_16x16x4_f32(false, a, false, bn,
                                                         (short)0, accn, false, false);
        }
        // D layout: VGPR v -> row v + 8*hi ; lane%16 -> item.  Dot with h in-register.
        #pragma unroll
        for (int v = 0; v < 8; ++v) {
            int d = rb * 16 + v + 8 * hi;
            float hv = sh_h[d * LSTRIDE + m];
            sp += accp[v] * hv;
            sn += accn[v] * hv;
        }
    }
    // fold the two half-waves (rows d and d+8 partials live in lanes n and n+16)
    sp += __shfl_xor(sp, 16, 32);
    sn += __shfl_xor(sn, 16, 32);
    if (lane < 16) { sh_sp[wave][m] = sp; sh_sn[wave][m] = sn; }
    __syncthreads();

    // ---- finalize per item: fixed-order sum -> stable softplus loss + L2 ----
    if (tid < cnt) {
        int i = sorted[start + tid];
        float sP = 0.f, sN = 0.f;
        #pragma unroll
        for (int w = 0; w < 4; ++w) { sP += sh_sp[w][tid]; sN += sh_sn[w][tid]; }
        float l2 = 0.f;
        for (int k = 0; k < DIM; ++k) {
            float a = sh_h[k * LSTRIDE + tid];
            float b = sh_tp[k * LSTRIDE + tid];
            float c = sh_tn[k * LSTRIDE + tid];
            l2 += a * a + b * b + c * c;
        }
        // pos_score = -sP, neg_score = -sN  =>  z = neg - pos = sP - sN
        float z  = sP - sN;
        float lb = (z > 0.f) ? log1pf(expf(-z)) : (-z + log1pf(expf(z)));  // -log sigmoid(z)
        loss[i]  = lb + 1e-5f * 0.5f * (l2 + relnorm[g]);
    }
}

// fixed-order final reduction -> scalar mean
__global__ void k_reduce(const float* __restrict__ loss, float* __restrict__ out) {
    __shared__ float red[256];
    float s = 0.f;
    for (int i = threadIdx.x; i < BATCH; i += 256) s += loss[i];
    red[threadIdx.x] = s;
    __syncthreads();
    for (int off = 128; off > 0; off >>= 1) {
        if (threadIdx.x < off) red[threadIdx.x] += red[threadIdx.x + off];
        __syncthreads();
    }
    if (threadIdx.x == 0) out[0] = red[0] * (1.0f / (float)BATCH);
}

// ---------------- launch ----------------

extern "C" void kernel_launch(void* const* d_in, const int* in_sizes, int n_in,
                              void* d_out, int out_size, void* d_ws, size_t ws_size,
                              hipStream_t stream) {
    const int*   h     = (const int*)d_in[0];
    const int*   r     = (const int*)d_in[1];
    const int*   pos_t = (const int*)d_in[2];
    const int*   neg_t = (const int*)d_in[3];
    const float* ent   = (const float*)d_in[4];
    const float* rel   = (const float*)d_in[5];

    int*   counts  = (int*)d_ws;               // 200 (padded)
    int*   offsets = counts + 256;             // 200 (padded)
    int*   ntiles  = offsets + 256;            // 1  (padded)
    int*   tg      = ntiles + 64;              // MAXTILES
    int*   ts      = tg + MAXTILES;            // MAXTILES
    int*   tc      = ts + MAXTILES;            // MAXTILES
    int*   sorted  = tc + MAXTILES;            // BATCH
    float* relnorm = (float*)(sorted + BATCH); // 200 (padded)
    float* loss    = relnorm + 256;            // BATCH

    k_zero   <<<1, 256, 0, stream>>>(counts, ntiles);
    k_relnorm<<<NREL, 256, 0, stream>>>(rel, relnorm);
    k_hist   <<<BATCH / 256, 256, 0, stream>>>(r, counts);
    k_scan   <<<1, 32, 0, stream>>>(counts, offsets, tg, ts, tc, ntiles);
    k_scatter<<<BATCH / 256, 256, 0, stream>>>(r, offsets, sorted);
    k_score  <<<MAXTILES, 128, 0, stream>>>(h, pos_t, neg_t, ent, rel,
                                            tg, ts, tc, ntiles, sorted, relnorm, loss);
    k_reduce <<<1, 256, 0, stream>>>(loss, (float*)d_out);
}